// UpsampleIntoPadding_11158325035230
// MI455X (gfx1250) — compile-verified
//
#include <hip/hip_runtime.h>

// Problem constants (match reference): B=32, NY=32, NX=32, D=1024
#define BB   32
#define NYY  32
#define NXX  32
#define DD   1024
#define SXS  1028   // sX row stride in floats; 1028 % 64 == 4 -> conflict-free B-frag reads
#define WSS  33     // sW row stride in floats; odd stride -> conflict-free A-frag reads
#define NTHREADS 256

typedef __attribute__((ext_vector_type(2))) float v2f;
typedef __attribute__((ext_vector_type(8))) float v8f;
typedef int v4i __attribute__((vector_size(16)));   // matches builtin param type

#define AS1 __attribute__((address_space(1)))
#define AS3 __attribute__((address_space(3)))

#ifndef __has_builtin
#define __has_builtin(x) 0
#endif

#if defined(__gfx1250__) && \
    __has_builtin(__builtin_amdgcn_global_load_async_to_lds_b128) && \
    __has_builtin(__builtin_amdgcn_s_wait_asynccnt)
#define USE_ASYNC_LDS 1
#else
#define USE_ASYNC_LDS 0
#if defined(__gfx1250__)
#warning "gfx1250 async-to-LDS builtins unavailable; using VGPR round-trip staging fallback"
#endif
#endif

// One block per b' = b*NX + nx. Stage the L valid source rows into LDS once
// (async global->LDS copy, ASYNCcnt-tracked, no VGPR round trip), build the
// 32x32 interpolation matrix W in LDS, then compute out = W @ Xsrc with
// fp32 WMMA (V_WMMA_F32_16X16X4_F32) tiles.
__global__ __launch_bounds__(NTHREADS)
void upsample_into_padding_wmma(const float* __restrict__ x,
                                const int*   __restrict__ mask,
                                float*       __restrict__ out)
{
    __shared__ __align__(16) float sX[NYY * SXS];   // ~128.5 KB staged source rows
    __shared__ float sW[NYY * WSS];                 // 32x32 interpolation weights (padded)
    __shared__ int   sL;

    const int tid = threadIdx.x;
    const int bp  = blockIdx.x;        // b' in [0, B*NX)
    const int b   = bp >> 5;           // bp / NX
    const int nx  = bp & 31;           // bp % NX

    // ---- zero W, compute valid prefix length L ----
    for (int i = tid; i < NYY * WSS; i += NTHREADS) sW[i] = 0.0f;
    if (tid == 0) {
        int L = 0;
        for (int y = 0; y < NYY; ++y)
            L += mask[(b * NYY + y) * NXX + nx];   // mask[b, y, nx]
        sL = L;
    }
    __syncthreads();

    const int L = sL;                  // uniform across block, L >= 1

    // ---- stage source rows into LDS: each lane copies its own 16B ----
    // x[b, y, nx, :] is 1024 contiguous floats = 256 x 16B per row.
#if USE_ASYNC_LDS
    for (int y = 0; y < L; ++y) {
        const float* g = x + (((size_t)b * NYY + y) * NXX + nx) * DD + tid * 4;
        float*       l = &sX[y * SXS + tid * 4];
        __builtin_amdgcn_global_load_async_to_lds_b128(
            (AS1 v4i*)g, (AS3 v4i*)l, /*imm offset=*/0, /*cpol=*/0);
    }
#else
    for (int y = 0; y < L; ++y) {
        const float4 v = reinterpret_cast<const float4*>(
            x + (((size_t)b * NYY + y) * NXX + nx) * DD)[tid];
        reinterpret_cast<float4*>(&sX[y * SXS])[tid] = v;
    }
#endif

    // ---- interpolation weights (same float sequencing as reference),
    //      overlapped with the in-flight async copies ----
    if (tid < NYY) {
        const int   t   = tid;
        const float Lf  = (float)L;
        float src = ((float)t + 0.5f) * (Lf / 32.0f) - 0.5f;
        src = fminf(fmaxf(src, 0.0f), Lf - 1.0f);
        const int   lo = (int)floorf(src);
        const int   hi = min(lo + 1, L - 1);
        const float w  = src - (float)lo;
        // row t owned by one thread -> plain accumulate (handles lo==hi, w==0)
        sW[t * WSS + lo] += 1.0f - w;
        sW[t * WSS + hi] += w;
    }

    // ---- zero-fill rows >= L so 0-weight columns can't inject NaN ----
    const float4 z4 = make_float4(0.0f, 0.0f, 0.0f, 0.0f);
    for (int y = L; y < NYY; ++y) {
        reinterpret_cast<float4*>(&sX[y * SXS])[tid] = z4;
    }

#if USE_ASYNC_LDS
    __builtin_amdgcn_s_wait_asynccnt(0);   // this wave's async copies landed in LDS
#endif
    __syncthreads();

    // ---- out(32 x 1024) = W(32 x 32) @ Xsrc(32 x 1024) via fp32 WMMA ----
    // 2 M-tiles x 64 N-tiles = 128 output tiles; wave w owns a CONTIGUOUS
    // band of 8 N-tiles per M-tile -> its 64B store segments merge into
    // full L2 lines.
    const int wave = tid >> 5;
    const int lane = tid & 31;
    const int n    = lane & 15;        // column within tile (A: row m == n index)
    const int half = lane >> 4;        // lane half selects K pair / C row offset

    for (int j = 0; j < 16; ++j) {
        const int mt = j >> 3;                 // 0..1
        const int nt = wave * 8 + (j & 7);     // contiguous 8-tile band per wave

        v8f acc = {};
        const int arow = (mt * 16 + n) * WSS;   // A: lane m = lane&15
        const int bcol = nt * 16 + n;           // B: lane n = lane&15

        #pragma unroll
        for (int k0 = 0; k0 < 32; k0 += 4) {
            const int k = k0 + 2 * half;
            v2f afrag, bfrag;
            afrag.x = sW[arow + k];
            afrag.y = sW[arow + k + 1];
            bfrag.x = sX[k * SXS + bcol];
            bfrag.y = sX[(k + 1) * SXS + bcol];
            acc = __builtin_amdgcn_wmma_f32_16x16x4_f32(
                /*neg_a=*/false, afrag, /*neg_b=*/false, bfrag,
                /*c_mod=*/(short)0, acc, /*reuse_a=*/false, /*reuse_b=*/false);
        }

        // C/D layout: VGPR i -> row (mt*16 + i + 8*half), col (nt*16 + n)
        float* orow = out + ((size_t)bp * NYY + mt * 16 + 8 * half) * DD + nt * 16 + n;
        #pragma unroll
        for (int i = 0; i < 8; ++i)
            orow[(size_t)i * DD] = acc[i];
    }
}

extern "C" void kernel_launch(void* const* d_in, const int* in_sizes, int n_in,
                              void* d_out, int out_size, void* d_ws, size_t ws_size,
                              hipStream_t stream) {
    const float* x    = (const float*)d_in[0];   // (32, 1024, 1024) fp32
    const int*   mask = (const int*)d_in[1];     // (32, 1024) int32
    float*       out  = (float*)d_out;           // (1024, 32, 1024) fp32

    dim3 grid(BB * NXX);    // 1024 blocks, one per (b, nx)
    dim3 block(NTHREADS);   // 8 waves (wave32)
    upsample_into_padding_wmma<<<grid, block, 0, stream>>>(x, mask, out);
}